// AttentionDecoder_42872363548874
// MI455X (gfx1250) — compile-verified
//
#include <hip/hip_runtime.h>
#include <hip/hip_bf16.h>
#include <math.h>

#define H 1024
#define V 50257
#define S 20

typedef __attribute__((ext_vector_type(2))) float v2f;
typedef __attribute__((ext_vector_type(8))) float v8f;

// ---------------------------------------------------------------- K0: zero scores
__global__ void k_init(float* __restrict__ scores) {
    if (threadIdx.x < 32) scores[threadIdx.x] = 0.0f;
}

// ---------------------------------------------------------------- K1: attention scores via WMMA f32 16x16x4
// A = attn_w[1024x2048] @ X[2048x32], X[:,s] = [hidden ; enc_s] (cols >= 20 zero-masked),
// then scores[s] += v_w[m] * tanh(A[m,s] + attn_b[m]).
// Branch-free K loops; 2 independent accumulators break the WMMA->WMMA RAW chain so the
// scheduler can overlap the two load/XDL streams.
__global__ __launch_bounds__(256) void k_attn_scores(
    const float* __restrict__ hidden, const float* __restrict__ enc,
    const float* __restrict__ attn_w, const float* __restrict__ attn_b,
    const float* __restrict__ v_w, float* __restrict__ scores)
{
    const int lane  = threadIdx.x & 31;
    const int wid   = (int)((blockIdx.x * blockDim.x + threadIdx.x) >> 5); // 0..127
    const int mtile = wid >> 1;           // 0..63
    const int ntile = wid & 1;            // 0..1
    const int l16   = lane & 15;
    const int half  = lane >> 4;          // selects K pair {0,1} vs {2,3}
    const int arow  = mtile * 16 + l16;   // A row (M)
    const int col   = ntile * 16 + l16;   // B/C column (source position s)
    const bool inS  = (col < S);
    const float bmask = inS ? 1.0f : 0.0f;

    const float2* __restrict__ w2 = (const float2*)(attn_w + (size_t)arow * (2 * H));
    const float2* __restrict__ h2 = (const float2*)hidden;
    const float2* __restrict__ e2 = (const float2*)(enc + (size_t)(inS ? col : 0) * H);

    v8f acc0 = {0.f, 0.f, 0.f, 0.f, 0.f, 0.f, 0.f, 0.f};
    v8f acc1 = {0.f, 0.f, 0.f, 0.f, 0.f, 0.f, 0.f, 0.f};

    // Phase 1: k in [0, H) — B from hidden (identical for all columns)
    for (int p = half; p < H / 2; p += 4) {
        const float2 av0 = w2[p];
        const float2 bv0 = h2[p];
        const float2 av1 = w2[p + 2];
        const float2 bv1 = h2[p + 2];
        v2f a0; a0.x = av0.x; a0.y = av0.y;
        v2f b0; b0.x = bv0.x; b0.y = bv0.y;
        v2f a1; a1.x = av1.x; a1.y = av1.y;
        v2f b1; b1.x = bv1.x; b1.y = bv1.y;
        acc0 = __builtin_amdgcn_wmma_f32_16x16x4_f32(
            false, a0, false, b0, (short)0, acc0, false, false);
        acc1 = __builtin_amdgcn_wmma_f32_16x16x4_f32(
            false, a1, false, b1, (short)0, acc1, false, false);
    }
    // Phase 2: k in [H, 2H) — B from enc[col], zero-masked for padded columns
    for (int p = half; p < H / 2; p += 4) {
        const float2 av0 = w2[H / 2 + p];
        const float2 ev0 = e2[p];
        const float2 av1 = w2[H / 2 + p + 2];
        const float2 ev1 = e2[p + 2];
        v2f a0; a0.x = av0.x; a0.y = av0.y;
        v2f b0; b0.x = bmask * ev0.x; b0.y = bmask * ev0.y;
        v2f a1; a1.x = av1.x; a1.y = av1.y;
        v2f b1; b1.x = bmask * ev1.x; b1.y = bmask * ev1.y;
        acc0 = __builtin_amdgcn_wmma_f32_16x16x4_f32(
            false, a0, false, b0, (short)0, acc0, false, false);
        acc1 = __builtin_amdgcn_wmma_f32_16x16x4_f32(
            false, a1, false, b1, (short)0, acc1, false, false);
    }

    // C layout: VGPR r, lanes 0-15 -> M = r ; lanes 16-31 -> M = r + 8
    float partial = 0.0f;
    #pragma unroll
    for (int r = 0; r < 8; ++r) {
        const int m = mtile * 16 + r + 8 * half;
        const float e = tanhf(acc0[r] + acc1[r] + attn_b[m]);
        partial += v_w[m] * e;
    }
    if (inS) atomicAdd(&scores[col], partial);
}

// ---------------------------------------------------------------- K2: softmax(20) + weighted context
__global__ __launch_bounds__(256) void k_softmax_weighted(
    const float* __restrict__ enc, const float* __restrict__ scores,
    float* __restrict__ weighted)
{
    __shared__ float a[S];
    if (threadIdx.x < S) {
        float m = -INFINITY;
        for (int s = 0; s < S; ++s) m = fmaxf(m, scores[s]);
        float sum = 0.0f;
        for (int s = 0; s < S; ++s) sum += __expf(scores[s] - m);
        a[threadIdx.x] = __expf(scores[threadIdx.x] - m) / sum;
    }
    __syncthreads();
    for (int h = threadIdx.x; h < H; h += blockDim.x) {
        float acc = 0.0f;
        #pragma unroll
        for (int s = 0; s < S; ++s) acc += a[s] * enc[s * H + h];
        weighted[h] = acc;
    }
}

// ---------------------------------------------------------------- K3: single-step GRU, wave per output row
__global__ __launch_bounds__(256) void k_gru(
    const int* __restrict__ token, const float* __restrict__ hidden,
    const float* __restrict__ embed_table,
    const float* __restrict__ w_ih, const float* __restrict__ w_hh,
    const float* __restrict__ b_ih, const float* __restrict__ b_hh,
    const float* __restrict__ weighted, float* __restrict__ h_new)
{
    const int lane = threadIdx.x & 31;
    const int j = (int)((blockIdx.x * blockDim.x + threadIdx.x) >> 5);
    if (j >= H) return;

    const float4* __restrict__ e4 = (const float4*)(embed_table + (size_t)token[0] * H);
    const float4* __restrict__ c4 = (const float4*)weighted;
    const float4* __restrict__ h4 = (const float4*)hidden;
    const float4* __restrict__ wr = (const float4*)(w_ih + (size_t)j * 2 * H);
    const float4* __restrict__ wz = (const float4*)(w_ih + (size_t)(j + H) * 2 * H);
    const float4* __restrict__ wn = (const float4*)(w_ih + (size_t)(j + 2 * H) * 2 * H);
    const float4* __restrict__ ur = (const float4*)(w_hh + (size_t)j * H);
    const float4* __restrict__ uz = (const float4*)(w_hh + (size_t)(j + H) * H);
    const float4* __restrict__ un = (const float4*)(w_hh + (size_t)(j + 2 * H) * H);

    float sr = 0.f, sz = 0.f, sn = 0.f, tr = 0.f, tz = 0.f, tn = 0.f;

    for (int q = lane; q < H / 4; q += 32) {             // x[0:H) = embed
        float4 x = e4[q];
        float4 a = wr[q], b = wz[q], c = wn[q];
        sr += x.x*a.x + x.y*a.y + x.z*a.z + x.w*a.w;
        sz += x.x*b.x + x.y*b.y + x.z*b.z + x.w*b.w;
        sn += x.x*c.x + x.y*c.y + x.z*c.z + x.w*c.w;
    }
    for (int q = lane; q < H / 4; q += 32) {             // x[H:2H) = context
        float4 x = c4[q];
        float4 a = wr[q + H/4], b = wz[q + H/4], c = wn[q + H/4];
        sr += x.x*a.x + x.y*a.y + x.z*a.z + x.w*a.w;
        sz += x.x*b.x + x.y*b.y + x.z*b.z + x.w*b.w;
        sn += x.x*c.x + x.y*c.y + x.z*c.z + x.w*c.w;
    }
    for (int q = lane; q < H / 4; q += 32) {             // hidden
        float4 x = h4[q];
        float4 a = ur[q], b = uz[q], c = un[q];
        tr += x.x*a.x + x.y*a.y + x.z*a.z + x.w*a.w;
        tz += x.x*b.x + x.y*b.y + x.z*b.z + x.w*b.w;
        tn += x.x*c.x + x.y*c.y + x.z*c.z + x.w*c.w;
    }
    #pragma unroll
    for (int off = 16; off > 0; off >>= 1) {
        sr += __shfl_down(sr, off, 32);
        sz += __shfl_down(sz, off, 32);
        sn += __shfl_down(sn, off, 32);
        tr += __shfl_down(tr, off, 32);
        tz += __shfl_down(tz, off, 32);
        tn += __shfl_down(tn, off, 32);
    }
    if (lane == 0) {
        const float gir = sr + b_ih[j],         ghr = tr + b_hh[j];
        const float giz = sz + b_ih[j + H],     ghz = tz + b_hh[j + H];
        const float gin = sn + b_ih[j + 2 * H], ghn = tn + b_hh[j + 2 * H];
        const float r = 1.0f / (1.0f + __expf(-(gir + ghr)));
        const float z = 1.0f / (1.0f + __expf(-(giz + ghz)));
        const float n = tanhf(gin + r * ghn);
        h_new[j] = (1.0f - z) * n + z * hidden[j];
    }
}

// ---------------------------------------------------------------- K4: output projection GEMV (bandwidth dominant)
__global__ __launch_bounds__(256) void k_logits(
    const float* __restrict__ out_w, const float* __restrict__ out_b,
    const float* __restrict__ h_new, float* __restrict__ logits)
{
    const int lane = threadIdx.x & 31;
    const int v = (int)((blockIdx.x * blockDim.x + threadIdx.x) >> 5);
    if (v >= V) return;
    const float4* __restrict__ w4 = (const float4*)(out_w + (size_t)v * H);
    const float4* __restrict__ h4 = (const float4*)h_new;
    float s = 0.0f;
    #pragma unroll
    for (int q = lane; q < H / 4; q += 32) {
        float4 w = w4[q];
        float4 h = h4[q];
        s += w.x*h.x + w.y*h.y + w.z*h.z + w.w*h.w;
    }
    #pragma unroll
    for (int off = 16; off > 0; off >>= 1) s += __shfl_down(s, off, 32);
    if (lane == 0) logits[v] = s + out_b[v];
}

// ---------------------------------------------------------------- K5: logsumexp over V (single block)
__global__ __launch_bounds__(1024) void k_lse(
    const float* __restrict__ logits, float* __restrict__ lse)
{
    __shared__ float red[1024];
    float m = -INFINITY;
    for (int v = threadIdx.x; v < V; v += 1024) m = fmaxf(m, logits[v]);
    red[threadIdx.x] = m;
    __syncthreads();
    for (int off = 512; off > 0; off >>= 1) {
        if (threadIdx.x < off)
            red[threadIdx.x] = fmaxf(red[threadIdx.x], red[threadIdx.x + off]);
        __syncthreads();
    }
    m = red[0];
    __syncthreads();
    float s = 0.0f;
    for (int v = threadIdx.x; v < V; v += 1024) s += __expf(logits[v] - m);
    red[threadIdx.x] = s;
    __syncthreads();
    for (int off = 512; off > 0; off >>= 1) {
        if (threadIdx.x < off) red[threadIdx.x] += red[threadIdx.x + off];
        __syncthreads();
    }
    if (threadIdx.x == 0) lse[0] = m + logf(red[0]);
}

// ---------------------------------------------------------------- K6: log_softmax finalize (in-place)
__global__ void k_logsoftmax(float* __restrict__ pred, const float* __restrict__ lse) {
    const int v = blockIdx.x * blockDim.x + threadIdx.x;
    if (v < V) pred[v] -= lse[0];
}

// ----------------------------------------------------------------
extern "C" void kernel_launch(void* const* d_in, const int* in_sizes, int n_in,
                              void* d_out, int out_size, void* d_ws, size_t ws_size,
                              hipStream_t stream) {
    const int*   token   = (const int*)d_in[0];
    const float* hidden  = (const float*)d_in[1];
    const float* enc     = (const float*)d_in[2];
    const float* embed   = (const float*)d_in[3];
    const float* attn_w  = (const float*)d_in[4];
    const float* attn_b  = (const float*)d_in[5];
    const float* v_w     = (const float*)d_in[6];
    const float* w_ih    = (const float*)d_in[7];
    const float* w_hh    = (const float*)d_in[8];
    const float* b_ih    = (const float*)d_in[9];
    const float* b_hh    = (const float*)d_in[10];
    const float* out_w   = (const float*)d_in[11];
    const float* out_b   = (const float*)d_in[12];

    float* out = (float*)d_out;
    float* ws  = (float*)d_ws;

    float* scores   = ws;          // 32 floats
    float* weighted = ws + 64;     // 1024 floats
    float* lse      = ws + 2048;   // 1 float
    float* logits   = out;         // V floats (prediction, finalized in place)
    float* h_new    = out + V;     // H floats (second output)

    k_init<<<1, 64, 0, stream>>>(scores);
    k_attn_scores<<<16, 256, 0, stream>>>(hidden, enc, attn_w, attn_b, v_w, scores);
    k_softmax_weighted<<<1, 256, 0, stream>>>(enc, scores, weighted);
    k_gru<<<(H * 32) / 256, 256, 0, stream>>>(token, hidden, embed, w_ih, w_hh,
                                              b_ih, b_hh, weighted, h_new);
    k_logits<<<(V + 7) / 8, 256, 0, stream>>>(out_w, out_b, h_new, logits);
    k_lse<<<1, 1024, 0, stream>>>(logits, lse);
    k_logsoftmax<<<(V + 255) / 256, 256, 0, stream>>>(logits, lse);
}